// EuclideanBlock_27324581937366
// MI455X (gfx1250) — compile-verified
//
#include <hip/hip_runtime.h>
#include <hip/hip_bf16.h>

// dist[b, s*hw + p] = sqrt( sum_c (x1[b,c,p] - x2[b,s,c,p])^2 )
// B=32, S=25, C=64, hw=42*42=1764.  Memory-bound (AI ~0.7 flop/byte):
// stream x2 once at HBM rate, x1 re-reads served by L2.
//
// WMMA use: A[m=p(16 lanes), k=c(4)] = diff^2, B = ones -> D[m,n] = sum_k A[m,k],
// C-accumulated over 16 chained V_WMMA_F32_16X16X4_F32 ops = full 64-channel sum.

#define B_DIM 32
#define S_DIM 25
#define C_DIM 64
#define HW    1764   // 42*42
#define PTILE 128    // 8 waves * 16 p per block
#define NTILE 14     // ceil(1764/128)

typedef __attribute__((ext_vector_type(2))) float v2f;
typedef __attribute__((ext_vector_type(8))) float v8f;

#if !__has_builtin(__builtin_amdgcn_wmma_f32_16x16x4_f32)
#warning "gfx1250: __builtin_amdgcn_wmma_f32_16x16x4_f32 not available; falling back to VALU reduction"
#define EUCL_HAVE_WMMA_F32 0
#else
#define EUCL_HAVE_WMMA_F32 1
#endif

__global__ __launch_bounds__(256) void EuclideanBlock_wmma_kernel(
    const float* __restrict__ x1,   // [B, C, hw]
    const float* __restrict__ x2,   // [B, S, C, hw]
    float* __restrict__ out)        // [B, S*hw]
{
    const int tile = blockIdx.x;            // 0..13
    const int s    = blockIdx.y;            // 0..24
    const int b    = blockIdx.z;            // 0..31
    const int lane = threadIdx.x & 31;      // wave32
    const int wave = threadIdx.x >> 5;

    const int p_base = tile * PTILE + wave * 16;
    if (p_base >= HW) return;               // wave-uniform: EXEC stays all-ones

    // A-matrix placement for V_WMMA_F32_16X16X4_F32 (wave32):
    //   lanes 0-15 : row M = lane,     K = 0,1 in a.x, a.y
    //   lanes 16-31: row M = lane-16,  K = 2,3 in a.x, a.y
    const int row   = lane & 15;
    const int kbase = (lane >> 4) << 1;     // 0 or 2

    int p = p_base + row;
    if (p > HW - 1) p = HW - 1;             // clamp loads for the straddling wave

    const float* x1p = x1 + ((size_t)b * C_DIM + kbase) * HW + p;
    const float* x2p = x2 + (((size_t)b * S_DIM + s) * C_DIM + kbase) * HW + p;

#if EUCL_HAVE_WMMA_F32
    v8f acc = {};
    const v2f ones = {1.0f, 1.0f};
    #pragma unroll
    for (int c0 = 0; c0 < C_DIM; c0 += 4) {
        float d0 = x1p[0]  - x2p[0];        // channel kbase
        float d1 = x1p[HW] - x2p[HW];       // channel kbase+1
        v2f a;
        a.x = d0 * d0;
        a.y = d1 * d1;
        // D[m,n] += sum_k A[m,k] * 1  ->  per-row channel-sum, replicated over n
        acc = __builtin_amdgcn_wmma_f32_16x16x4_f32(
            /*neg_a=*/false, a, /*neg_b=*/false, ones,
            /*c_mod=*/(short)0, acc, /*reuse_a=*/false, /*reuse_b=*/false);
        x1p += 4 * HW;
        x2p += 4 * HW;
    }
    // D layout: lane n in 0-15 holds rows 0..7 in acc[0..7]; lanes 16-31 rows 8..15.
    // All columns identical -> lanes 0 and 16 write the 16 results.
    if (row == 0) {
        const int m_off = (lane >> 4) * 8;  // 0 for lane 0, 8 for lane 16
        float* op = out + ((size_t)b * S_DIM + s) * HW + p_base + m_off;
        #pragma unroll
        for (int r = 0; r < 8; ++r) {
            const int pp = p_base + m_off + r;
            if (pp < HW) op[r] = __builtin_sqrtf(acc[r]);
        }
    }
#else
    // VALU fallback: same data layout, per-lane partial sums over (kbase, kbase+1),
    // combine the two half-wave partials through LDS.
    __shared__ float partial[256];
    float sum = 0.0f;
    #pragma unroll
    for (int c0 = 0; c0 < C_DIM; c0 += 4) {
        float d0 = x1p[0]  - x2p[0];
        float d1 = x1p[HW] - x2p[HW];
        sum += d0 * d0 + d1 * d1;
        x1p += 4 * HW;
        x2p += 4 * HW;
    }
    partial[threadIdx.x] = sum;
    __builtin_amdgcn_s_barrier();
    if (lane < 16) {
        const int pp = p_base + row;
        if (pp < HW) {
            float tot = sum + partial[threadIdx.x + 16];
            out[((size_t)b * S_DIM + s) * HW + pp] = __builtin_sqrtf(tot);
        }
    }
#endif
}

extern "C" void kernel_launch(void* const* d_in, const int* in_sizes, int n_in,
                              void* d_out, int out_size, void* d_ws, size_t ws_size,
                              hipStream_t stream) {
    const float* x1 = (const float*)d_in[0];   // [32, 64, 42, 42]
    const float* x2 = (const float*)d_in[1];   // [32, 25, 64, 42, 42]
    float* out = (float*)d_out;                // [32, 25*42*42]
    (void)in_sizes; (void)n_in; (void)out_size; (void)d_ws; (void)ws_size;

    dim3 grid(NTILE, S_DIM, B_DIM);            // (14, 25, 32)
    dim3 block(256);
    hipLaunchKernelGGL(EuclideanBlock_wmma_kernel, grid, block, 0, stream,
                       x1, x2, out);
}